// RepresentationNetwork_84980222918908
// MI455X (gfx1250) — compile-verified
//
#include <hip/hip_runtime.h>
#include <hip/hip_bf16.h>
#include <stdint.h>

typedef __attribute__((ext_vector_type(2))) float v2f;
typedef __attribute__((ext_vector_type(8))) float v8f;

#define FDIM 128
#define MROWS 64          // rows per block (4 x 16-row WMMA tiles per wave)
#define LDS_PITCH 132     // bank-conflict-free pitch: bank = (4m + k) % 64

// ---------------------------------------------------------------------------
// Zero the aggregation buffer (float4 vectorized).
// ---------------------------------------------------------------------------
__global__ __launch_bounds__(256) void zero_f4(float4* __restrict__ p, int n4) {
    int i = blockIdx.x * 256 + threadIdx.x;
    if (i < n4) p[i] = make_float4(0.f, 0.f, 0.f, 0.f);
}

// ---------------------------------------------------------------------------
// Edge scatter: agg[dst] += h[src].  One wave32 per edge; each lane moves a
// float4 (32 lanes * 16B = 512B = one 128-float row).  Native fp32 global
// atomics — the 25.6MB node table lives entirely in the 192MB L2.
// ---------------------------------------------------------------------------
__global__ __launch_bounds__(256) void edge_scatter(
    const float* __restrict__ h,
    const long long* __restrict__ src,
    const long long* __restrict__ dst,
    float* __restrict__ agg,
    int n_edges)
{
    int e = (blockIdx.x * 256 + threadIdx.x) >> 5;   // 8 edges per block
    if (e >= n_edges) return;
    int lane = threadIdx.x & 31;

    long long s = src[e];
    long long d = dst[e];

    const float4 v = *(const float4*)(h + (size_t)s * FDIM + lane * 4);
    float* o = agg + (size_t)d * FDIM + lane * 4;
    unsafeAtomicAdd(o + 0, v.x);
    unsafeAtomicAdd(o + 1, v.y);
    unsafeAtomicAdd(o + 2, v.z);
    unsafeAtomicAdd(o + 3, v.w);
}

// ---------------------------------------------------------------------------
// Fused GraphConv GEMM:  hout = relu(agg @ w_rel^T + b_rel + hin @ w_root^T)
//
// Block = 256 threads (8 waves) = 64 node rows x all 128 output columns.
// A tiles (agg + hin, 64x128 each) staged in LDS with pitch 132 so the
// 16-lane fragment reads hit 64 distinct banks.  Each wave owns output
// columns [16w,16w+16) and computes 4 stacked 16x16 fp32 C tiles, so one
// B (weight) fragment load feeds 4 v_wmma_f32_16x16x4_f32 ops.
//
// fp32 WMMA operand mapping (ISA 7.12.2):
//   A: lane -> { A[m][k0], A[m][k0+1] },  m = lane&15, k0 = (lane>>4)*2
//   B: lane -> { B[k0][n], B[k0+1][n] },  n = lane&15   (B[k][n] = W[n][k])
//   C: vgpr r, lane -> C[r + (lane>>4)*8][lane&15]
// ---------------------------------------------------------------------------
__global__ __launch_bounds__(256) void graphconv_gemm(
    const float* __restrict__ agg,
    const float* __restrict__ hin,
    const float* __restrict__ w_rel,
    const float* __restrict__ w_root,
    const float* __restrict__ b_rel,
    float* __restrict__ hout,
    int n_nodes)
{
    __shared__ float sA[MROWS * LDS_PITCH];   // agg tile
    __shared__ float sX[MROWS * LDS_PITCH];   // hin tile

    const int tid  = threadIdx.x;
    const int lane = tid & 31;
    const int wave = tid >> 5;
    const int row0 = blockIdx.x * MROWS;

    // Cooperative stage: 64 rows x 128 floats per matrix (8 float4 / thread).
    // Clamp row index for the partial last block (reads stay in-bounds; the
    // duplicated rows are never stored).
    for (int i = tid; i < MROWS * (FDIM / 4); i += 256) {
        int r  = i >> 5;            // 32 float4 per row
        int c  = (i & 31) * 4;
        int rg = row0 + r;
        if (rg >= n_nodes) rg = n_nodes - 1;
        const float4 va = *(const float4*)(agg + (size_t)rg * FDIM + c);
        const float4 vx = *(const float4*)(hin + (size_t)rg * FDIM + c);
        *(float4*)(sA + r * LDS_PITCH + c) = va;
        *(float4*)(sX + r * LDS_PITCH + c) = vx;
    }
    __syncthreads();

    const int m   = lane & 15;
    const int n   = wave * 16 + (lane & 15);   // output column
    const int k0  = (lane >> 4) * 2;           // K-pair selector

    v8f c0 = {}, c1 = {}, c2 = {}, c3 = {};

    const float* sArow = sA + m * LDS_PITCH + k0;
    const float* sXrow = sX + m * LDS_PITCH + k0;
    const float* b1row = w_rel  + (size_t)n * FDIM + k0;
    const float* b2row = w_root + (size_t)n * FDIM + k0;

    // GEMM 1: agg @ w_rel^T — one B load feeds 4 WMMAs.
#pragma unroll
    for (int k = 0; k < FDIM; k += 4) {
        v2f b  = *(const v2f*)(b1row + k);
        v2f a0 = *(const v2f*)(sArow + 0 * 16 * LDS_PITCH + k);
        v2f a1 = *(const v2f*)(sArow + 1 * 16 * LDS_PITCH + k);
        v2f a2 = *(const v2f*)(sArow + 2 * 16 * LDS_PITCH + k);
        v2f a3 = *(const v2f*)(sArow + 3 * 16 * LDS_PITCH + k);
        c0 = __builtin_amdgcn_wmma_f32_16x16x4_f32(false, a0, false, b, (short)0, c0, false, false);
        c1 = __builtin_amdgcn_wmma_f32_16x16x4_f32(false, a1, false, b, (short)0, c1, false, false);
        c2 = __builtin_amdgcn_wmma_f32_16x16x4_f32(false, a2, false, b, (short)0, c2, false, false);
        c3 = __builtin_amdgcn_wmma_f32_16x16x4_f32(false, a3, false, b, (short)0, c3, false, false);
    }

    // GEMM 2: hin @ w_root^T (accumulate into the same C tiles).
#pragma unroll
    for (int k = 0; k < FDIM; k += 4) {
        v2f b  = *(const v2f*)(b2row + k);
        v2f a0 = *(const v2f*)(sXrow + 0 * 16 * LDS_PITCH + k);
        v2f a1 = *(const v2f*)(sXrow + 1 * 16 * LDS_PITCH + k);
        v2f a2 = *(const v2f*)(sXrow + 2 * 16 * LDS_PITCH + k);
        v2f a3 = *(const v2f*)(sXrow + 3 * 16 * LDS_PITCH + k);
        c0 = __builtin_amdgcn_wmma_f32_16x16x4_f32(false, a0, false, b, (short)0, c0, false, false);
        c1 = __builtin_amdgcn_wmma_f32_16x16x4_f32(false, a1, false, b, (short)0, c1, false, false);
        c2 = __builtin_amdgcn_wmma_f32_16x16x4_f32(false, a2, false, b, (short)0, c2, false, false);
        c3 = __builtin_amdgcn_wmma_f32_16x16x4_f32(false, a3, false, b, (short)0, c3, false, false);
    }

    // Bias + ReLU + store.  C vgpr r -> row (t*16 + r + (lane>>4)*8), col n.
    const float bias = b_rel[n];
    const int   mhi  = (lane >> 4) * 8;
    v8f acc[4] = {c0, c1, c2, c3};
#pragma unroll
    for (int t = 0; t < 4; ++t) {
#pragma unroll
        for (int r = 0; r < 8; ++r) {
            int row = row0 + t * 16 + mhi + r;
            if (row < n_nodes) {
                float v = acc[t][r] + bias;
                v = v > 0.f ? v : 0.f;
                hout[(size_t)row * FDIM + n] = v;
            }
        }
    }
}

// ---------------------------------------------------------------------------
// Driver: 3 layers, each = zero(agg); scatter; fused gemm.
// ws layout: [agg | h1 | h2], 3 * 50000*128 floats = 76.8 MB.
// ---------------------------------------------------------------------------
extern "C" void kernel_launch(void* const* d_in, const int* in_sizes, int n_in,
                              void* d_out, int out_size, void* d_ws, size_t ws_size,
                              hipStream_t stream) {
    const float*     x     = (const float*)d_in[0];
    const long long* ei    = (const long long*)d_in[1];
    const float*     wrel1 = (const float*)d_in[2];
    const float*     brel1 = (const float*)d_in[3];
    const float*     wrt1  = (const float*)d_in[4];
    const float*     wrel2 = (const float*)d_in[5];
    const float*     brel2 = (const float*)d_in[6];
    const float*     wrt2  = (const float*)d_in[7];
    const float*     wrel3 = (const float*)d_in[8];
    const float*     brel3 = (const float*)d_in[9];
    const float*     wrt3  = (const float*)d_in[10];
    float*           out   = (float*)d_out;

    const int n_nodes = in_sizes[0] / FDIM;   // 50000
    const int n_edges = in_sizes[1] / 2;      // 600000
    const long long* src = ei;
    const long long* dst = ei + n_edges;

    const size_t node_elems = (size_t)n_nodes * FDIM;
    float* agg = (float*)d_ws;
    float* h1  = agg + node_elems;
    float* h2  = h1 + node_elems;

    const int n4          = (int)(node_elems / 4);
    const int zero_blocks = (n4 + 255) / 256;
    const int scat_blocks = (n_edges * 32 + 255) / 256;
    const int gemm_blocks = (n_nodes + MROWS - 1) / MROWS;   // 782

    const float* hin  = x;
    const float* WR[3] = {wrel1, wrel2, wrel3};
    const float* BR[3] = {brel1, brel2, brel3};
    const float* WO[3] = {wrt1,  wrt2,  wrt3};
    float*       HO[3] = {h1,    h2,    out};

    for (int l = 0; l < 3; ++l) {
        zero_f4<<<zero_blocks, 256, 0, stream>>>((float4*)agg, n4);
        edge_scatter<<<scat_blocks, 256, 0, stream>>>(hin, src, dst, agg, n_edges);
        graphconv_gemm<<<gemm_blocks, 256, 0, stream>>>(agg, hin, WR[l], WO[l], BR[l], HO[l], n_nodes);
        hin = HO[l];
    }
}